// Cell_35562329211103
// MI455X (gfx1250) — compile-verified
//
#include <hip/hip_runtime.h>
#include <math.h>

// ---------------------------------------------------------------------------
// SPDNet DARTS cell on gfx1250, eigendecomposition-free:
//   ReEig / logm / expm realized as Newton-Schulz + scaling-squaring matmul
//   chains; every 64x64 f32 matmul runs on V_WMMA_F32_16X16X4_F32 fed purely
//   from LDS.  Operands are staged global->LDS with the CDNA5 async-copy
//   engine (GLOBAL_LOAD_ASYNC_TO_LDS_B128 / ASYNCcnt); states written back
//   with GLOBAL_STORE_ASYNC_FROM_LDS_B128.
// One workgroup (8 wave32) per (b,c) SPD matrix; 512 workgroups.
// ---------------------------------------------------------------------------

typedef __attribute__((ext_vector_type(2))) float v2f;
typedef __attribute__((ext_vector_type(8))) float v8f;

#if defined(__HIP_DEVICE_COMPILE__) && !__has_builtin(__builtin_amdgcn_wmma_f32_16x16x4_f32)
#error "missing __builtin_amdgcn_wmma_f32_16x16x4_f32 on this toolchain"
#endif

#define NDIM      64
#define NELEM     4096          // 64*64 (global matrices, row stride 64)
#define LDST      68            // padded LDS row stride (floats).
                                // A-fragment lanes then hit banks (4m+c)%64:
                                // all 32 lanes distinct -> conflict-free.
#define NTHREADS  256           // 8 wave32

static constexpr float SPD_EPS       = 1e-4f;
static constexpr int   SQRT_ITERS    = 12;  // Newton-Schulz sqrt iterations
static constexpr int   LOG_STAGES    = 5;   // logm: # of inverse-scaling sqrts
static constexpr int   LOG_TERMS     = 8;   // logm: log(1+W) series terms
static constexpr int   EXP_SQUARINGS = 5;   // expm: scaling-squaring depth
static constexpr int   EXP_TERMS     = 9;   // expm: Taylor/Horner terms

// ------------------- async staging: global <-> LDS (CDNA5) -----------------
// 16 bytes per lane per issue; 4 issues cover a 64x64 f32 matrix.
// Global side is row stride 64; LDS side is padded row stride 68.
static __device__ __forceinline__ void stage_g2l(float* L, const float* G) {
  const int tid = threadIdx.x;
#pragma unroll
  for (int it = 0; it < 4; ++it) {
    const int chunk = tid + it * NTHREADS;    // 0..1023 16-byte chunks
    const int row   = chunk >> 4;
    const int c4    = (chunk & 15) << 2;      // float column
    const unsigned      loff = (unsigned)(size_t)(L + row * LDST + c4);
    const unsigned long long gp =
        (unsigned long long)(const void*)(G + row * NDIM + c4);
    asm volatile("global_load_async_to_lds_b128 %0, %1, off"
                 :
                 : "v"(loff), "v"(gp)
                 : "memory");
  }
  asm volatile("s_wait_asynccnt 0x0" ::: "memory");
  __syncthreads();
}

static __device__ __forceinline__ void store_l2g(float* G, const float* L) {
  const int tid = threadIdx.x;
#pragma unroll
  for (int it = 0; it < 4; ++it) {
    const int chunk = tid + it * NTHREADS;
    const int row   = chunk >> 4;
    const int c4    = (chunk & 15) << 2;
    const unsigned      loff = (unsigned)(size_t)(L + row * LDST + c4);
    const unsigned long long gp =
        (unsigned long long)(void*)(G + row * NDIM + c4);
    asm volatile("global_store_async_from_lds_b128 %0, %1, off"
                 :
                 : "v"(gp), "v"(loff)
                 : "memory");
  }
  asm volatile("s_wait_asynccnt 0x0" ::: "memory");
  __syncthreads();
}

// --------------------------- 64x64 matmul (WMMA) ---------------------------
// C = A * B (or A * B^T).  All operands LDS, row stride LDST.  Each wave
// computes two N-adjacent 16x16 tiles so the A fragment feeds two WMMAs.
// f32 WMMA operand layout (ISA 7.12.2):
//   A 16x4 : lanes 0-15 M=lane K={0,1}; lanes 16-31 M=lane-16 K={2,3}
//   B 4x16 : lanes 0-15 N=lane K={0,1}; lanes 16-31 N=lane-16 K={2,3}
//   C 16x16: VGPR r -> row r (lanes 0-15) / row r+8 (lanes 16-31), col=lane%16
template <bool TRANSB>
static __device__ __forceinline__ void mm64(float* Cm, const float* Am,
                                            const float* Bm) {
  const int tid  = threadIdx.x;
  const int lane = tid & 31;
  const int wave = tid >> 5;
  const int m    = lane & 15;          // M (for A) / N (for B) within tile
  const int kk   = (lane >> 4) << 1;   // K sub-pair: 0 or 2
  const int mt   = wave >> 1;          // tile row 0..3
  const int nt0  = (wave & 1) << 1;    // first tile col (0 or 2)

  v8f acc0 = {0.f, 0.f, 0.f, 0.f, 0.f, 0.f, 0.f, 0.f};
  v8f acc1 = {0.f, 0.f, 0.f, 0.f, 0.f, 0.f, 0.f, 0.f};

  for (int kb = 0; kb < 16; ++kb) {
    const int k0 = (kb << 2) + kk;
    v2f a, b0, b1;
    a.x = Am[(mt * 16 + m) * LDST + k0];
    a.y = Am[(mt * 16 + m) * LDST + k0 + 1];
    if (TRANSB) {
      // op(B)[k][n] = B[n][k]
      b0.x = Bm[(nt0 * 16 + m) * LDST + k0];
      b0.y = Bm[(nt0 * 16 + m) * LDST + k0 + 1];
      b1.x = Bm[((nt0 + 1) * 16 + m) * LDST + k0];
      b1.y = Bm[((nt0 + 1) * 16 + m) * LDST + k0 + 1];
    } else {
      b0.x = Bm[k0 * LDST + nt0 * 16 + m];
      b0.y = Bm[(k0 + 1) * LDST + nt0 * 16 + m];
      b1.x = Bm[k0 * LDST + (nt0 + 1) * 16 + m];
      b1.y = Bm[(k0 + 1) * LDST + (nt0 + 1) * 16 + m];
    }
    acc0 = __builtin_amdgcn_wmma_f32_16x16x4_f32(false, a, false, b0,
                                                 (short)0, acc0, false, false);
    acc1 = __builtin_amdgcn_wmma_f32_16x16x4_f32(false, a, false, b1,
                                                 (short)0, acc1, false, false);
  }

  const int mb = (lane >> 4) << 3;  // row offset 0 or 8 within tile
  float* c0 = Cm + (mt * 16 + mb) * LDST + nt0 * 16 + m;
  float* c1 = c0 + 16;
#pragma unroll
  for (int r = 0; r < 8; ++r) {
    c0[r * LDST] = acc0[r];
    c1[r * LDST] = acc1[r];
  }
  __syncthreads();
}

// ------------------------- elementwise 64x64 helpers -----------------------
// Iterate the 4096 logical elements; translate to padded LDS addresses.
// Every helper ends with a workgroup barrier.
static __device__ __forceinline__ bool is_diag(int i) {
  return (i >> 6) == (i & 63);
}
static __device__ __forceinline__ int padaddr(int i) {
  return (i >> 6) * LDST + (i & 63);
}
static __device__ __forceinline__ void ew_zero(float* D) {
  for (int i = threadIdx.x; i < NELEM; i += NTHREADS) D[padaddr(i)] = 0.f;
  __syncthreads();
}
static __device__ __forceinline__ void ew_copy(float* D, const float* S) {
  for (int i = threadIdx.x; i < NELEM; i += NTHREADS) {
    const int a = padaddr(i);
    D[a] = S[a];
  }
  __syncthreads();
}
static __device__ __forceinline__ void ew_scale(float* D, const float* S,
                                                float a) {
  for (int i = threadIdx.x; i < NELEM; i += NTHREADS) {
    const int p = padaddr(i);
    D[p] = a * S[p];
  }
  __syncthreads();
}
static __device__ __forceinline__ void ew_axpy(float* D, const float* S,
                                               float a) {
  for (int i = threadIdx.x; i < NELEM; i += NTHREADS) {
    const int p = padaddr(i);
    D[p] += a * S[p];
  }
  __syncthreads();
}
static __device__ __forceinline__ void ew_identity(float* D) {
  for (int i = threadIdx.x; i < NELEM; i += NTHREADS)
    D[padaddr(i)] = is_diag(i) ? 1.f : 0.f;
  __syncthreads();
}
// D = a*S + d*I
static __device__ __forceinline__ void ew_diag_shift(float* D, const float* S,
                                                     float a, float d) {
  for (int i = threadIdx.x; i < NELEM; i += NTHREADS) {
    const int p = padaddr(i);
    float v = a * S[p];
    if (is_diag(i)) v += d;
    D[p] = v;
  }
  __syncthreads();
}
// T = 1.5*I - 0.5*T   (Newton-Schulz update factor), in place
static __device__ __forceinline__ void ew_nsT(float* T) {
  for (int i = threadIdx.x; i < NELEM; i += NTHREADS) {
    const int p = padaddr(i);
    float v = -0.5f * T[p];
    if (is_diag(i)) v += 1.5f;
    T[p] = v;
  }
  __syncthreads();
}
// C = 0.5*C + 0.5*R + 0.5*eps*I   (ReEig reconstruction), in place on C
static __device__ __forceinline__ void ew_reig_fin(float* C, const float* R) {
  for (int i = threadIdx.x; i < NELEM; i += NTHREADS) {
    const int p = padaddr(i);
    float v = 0.5f * C[p] + 0.5f * R[p];
    if (is_diag(i)) v += 0.5f * SPD_EPS;
    C[p] = v;
  }
  __syncthreads();
}

static __device__ float trace64(const float* X, float* red) {
  const int tid = threadIdx.x;
  if (tid < 64) red[tid] = X[tid * (LDST + 1)];  // diag element i at i*(68+1)
  __syncthreads();
  if (tid == 0) {
    float s = 0.f;
    for (int i = 0; i < 64; ++i) s += red[i];
    red[0] = s;
  }
  __syncthreads();
  const float r = red[0];
  __syncthreads();
  return r;
}

// ------------------------- matrix-function kernels -------------------------
// sqrtm via coupled Newton-Schulz:  Y0=X/c, Z0=I;
//   T=(3I-ZY)/2; Y<-YT; Z<-TZ;  sqrt(X)=sqrt(c)*Y.
// In-place on X; S1,S2,S3 scratch (X joins the ping-pong set after init).
static __device__ void sqrtm_ns(float* X, float* S1, float* S2, float* S3,
                                float* red) {
  const float c = trace64(X, red) + 1e-20f;  // trace >= lambda_max for SPD
  ew_scale(S1, X, 1.f / c);                  // y
  ew_identity(S2);                           // z
  float* y = S1;
  float* z = S2;
  float* u = X;
  float* t = S3;
  for (int it = 0; it < SQRT_ITERS; ++it) {
    mm64<false>(t, z, y);  // t = z*y
    ew_nsT(t);             // t = 1.5I - 0.5 t
    mm64<false>(u, y, t);  // y' = y*t
    { float* p = y; y = u; u = p; }
    mm64<false>(u, t, z);  // z' = t*z
    { float* p = z; z = u; u = p; }
  }
  ew_scale(X, y, sqrtf(c));
}

// ReEig, in place on C:  C <- 0.5*[(C+eps I) + sqrtm((C-eps I)^2)]
static __device__ void reig(float* C, float* Y, float* Z, float* T, float* U,
                            float* red) {
  ew_diag_shift(Y, C, 1.f, -SPD_EPS);  // Y = C - eps I
  mm64<false>(Z, Y, Y);                // Z = (C-eps I)^2   (Y symmetric)
  sqrtm_ns(Z, Y, U, T, red);           // Z = |C - eps I|
  ew_reig_fin(C, Z);
}

// Acc += coef * logm(Src), via inverse scaling-and-squaring:
//   X <- Src^(1/2^s) (s NS sqrts), then log(X) = sum (-1)^(i+1) W^i / i,
//   W = X - I, scaled by 2^s.
static __device__ void logm_acc(float* Acc, const float* Src, float coef,
                                float* C, float* Y, float* Z, float* T,
                                float* red) {
  if (Src != C) ew_copy(C, Src);
  for (int s = 0; s < LOG_STAGES; ++s) sqrtm_ns(C, Y, Z, T, red);
  ew_diag_shift(T, C, 1.f, -1.f);  // T = W = X - I
  ew_copy(Y, T);                   // current power P = W
  float* p = Y;
  float* q = Z;
  const float a2 = coef * (float)(1 << LOG_STAGES);
  for (int i = 1; i <= LOG_TERMS; ++i) {
    const float ci = ((i & 1) ? 1.f : -1.f) / (float)i;
    ew_axpy(Acc, p, a2 * ci);
    if (i < LOG_TERMS) {
      mm64<false>(q, p, T);
      float* r = p; p = q; q = r;
    }
  }
}

// Dst = expm(Src), scaling-and-squaring with Horner Taylor.
static __device__ void expm(const float* Src, float* Dst, float* C, float* Y,
                            float* Z) {
  ew_scale(C, Src, 1.f / (float)(1 << EXP_SQUARINGS));
  ew_diag_shift(Y, C, 1.f / (float)EXP_TERMS, 1.f);  // R = I + X/n
  for (int k = EXP_TERMS - 1; k >= 1; --k) {
    mm64<false>(Z, C, Y);                      // Z = X*R
    ew_diag_shift(Y, Z, 1.f / (float)k, 1.f);  // R = I + Z/k
  }
  float* y = Y;
  float* z = Z;
  for (int s = 0; s < EXP_SQUARINGS; ++s) {
    mm64<false>(z, y, y);
    float* r = y; y = z; z = r;
  }
  if (Dst != y) ew_copy(Dst, y);
}

// ------------------------------- cell kernel -------------------------------
__global__ __launch_bounds__(NTHREADS) void spd_cell_kernel(
    const float* __restrict__ s0g, const float* __restrict__ s1g,
    const float* __restrict__ wts, const float* __restrict__ Wp0,
    const float* __restrict__ Wp1, const float* __restrict__ Wop,
    float* __restrict__ out, float* __restrict__ ws) {
  // LDS: 9 padded matrices (9 * 64*68*4B = 153KB) + reduction pad; within
  // the CDNA5 320KB/WGP budget (2 workgroups resident per WGP).
  __shared__ float C[64 * LDST], Y[64 * LDST], Z[64 * LDST];
  __shared__ float T[64 * LDST], U[64 * LDST];
  __shared__ float HB[64 * LDST];    // staged state h
  __shared__ float WB[64 * LDST];    // staged BiMap weight
  __shared__ float MIX[64 * LDST];   // per-edge log-barycenter accumulator
  __shared__ float CELL[64 * LDST];  // per-step log-barycenter accumulator
  __shared__ float red[64];

  const int bc = blockIdx.x;  // 0..511 == b*16 + c
  float* Sg = ws + (size_t)bc * 5 * NELEM;  // 5 DAG state slots (stride 64)

  // ---- preprocess: S0 = ReEig(Wp0 s0 Wp0^T), S1 = ReEig(Wp1 s1 Wp1^T) ----
  stage_g2l(HB, s0g + (size_t)bc * NELEM);
  stage_g2l(WB, Wp0);
  mm64<false>(Y, WB, HB);  // Y = W*X
  mm64<true >(C, Y, WB);   // C = (W*X)*W^T
  reig(C, Y, Z, T, U, red);
  store_l2g(Sg, C);

  stage_g2l(HB, s1g + (size_t)bc * NELEM);
  stage_g2l(WB, Wp1);
  mm64<false>(Y, WB, HB);
  mm64<true >(C, Y, WB);
  reig(C, Y, Z, T, U, red);
  store_l2g(Sg + NELEM, C);

  // ---------------------------- DARTS steps -------------------------------
  int offset  = 0;
  int nstates = 2;
  for (int step = 0; step < 3; ++step) {
    ew_zero(CELL);
    for (int j = 0; j < nstates; ++j) {
      const int e = offset + j;
      stage_g2l(HB, Sg + (size_t)j * NELEM);  // h resident in LDS for 4 uses
      ew_zero(MIX);
      // candidate 0: skip (h itself)
      logm_acc(MIX, HB, wts[e * 4 + 0], C, Y, Z, T, red);
      // candidates 1..3: ReEig(BiMap(h, W_ops[e][k]))
      for (int k = 1; k < 4; ++k) {
        stage_g2l(WB, Wop + ((size_t)e * 4 + k) * NELEM);
        mm64<false>(Y, WB, HB);
        mm64<true >(C, Y, WB);
        reig(C, Y, Z, T, U, red);
        logm_acc(MIX, C, wts[e * 4 + k], C, Y, Z, T, red);
      }
      expm(MIX, MIX, C, Y, Z);  // mixed_j (log-Euclidean barycenter)
      logm_acc(CELL, MIX, 1.f / (float)nstates, C, Y, Z, T, red);
    }
    expm(CELL, CELL, C, Y, Z);
    store_l2g(Sg + (size_t)(2 + step) * NELEM, CELL);
    offset += nstates;
    ++nstates;
  }

  // -------- output: concat(states[2..4]) along channels -> [B,3C,N,N] -----
  const int b = bc >> 4;
  const int c = bc & 15;
  for (int i = 0; i < 3; ++i) {
    const float* S = Sg + (size_t)(2 + i) * NELEM;
    float* dst = out + (((size_t)b * 48 + (size_t)(i * 16 + c)) * NELEM);
    for (int idx = threadIdx.x; idx < NELEM; idx += NTHREADS) dst[idx] = S[idx];
  }
}

// ------------------------------- host launch -------------------------------
extern "C" void kernel_launch(void* const* d_in, const int* in_sizes, int n_in,
                              void* d_out, int out_size, void* d_ws,
                              size_t ws_size, hipStream_t stream) {
  (void)in_sizes; (void)n_in; (void)out_size; (void)ws_size;
  const float* s0  = (const float*)d_in[0];
  const float* s1  = (const float*)d_in[1];
  const float* wts = (const float*)d_in[2];
  const float* Wp0 = (const float*)d_in[3];
  const float* Wp1 = (const float*)d_in[4];
  const float* Wop = (const float*)d_in[5];
  float* out = (float*)d_out;
  float* ws  = (float*)d_ws;  // needs 512 * 5 * 16KB = 40 MB

  hipLaunchKernelGGL(spd_cell_kernel, dim3(512), dim3(NTHREADS), 0, stream,
                     s0, s1, wts, Wp0, Wp1, Wop, out, ws);
}